// ChebConv_13288628814252
// MI455X (gfx1250) — compile-verified
//
#include <hip/hip_runtime.h>
#include <hip/hip_bf16.h>
#include <stdint.h>

#define IC 256
#define OC 256

typedef __attribute__((ext_vector_type(2))) float v2f;
typedef __attribute__((ext_vector_type(8))) float v8f;

#if defined(__has_builtin)
#if __has_builtin(__builtin_amdgcn_wmma_f32_16x16x4_f32)
#define HAVE_WMMA_F32 1
#endif
#endif

// ---------------- elementwise helpers ----------------

__global__ __launch_bounds__(256) void cheb_zero(float* __restrict__ p, int n) {
    int i = blockIdx.x * 256 + threadIdx.x;
    if (i < n) p[i] = 0.f;
}

// dst[i] = s * src[i], vectorized float4 (n4 = count of float4 elems). Safe in place.
__global__ __launch_bounds__(256) void cheb_scale_copy(float* __restrict__ dst,
                                                       const float* __restrict__ src,
                                                       float s, int n4) {
    int i = blockIdx.x * 256 + threadIdx.x;
    if (i < n4) {
        float4 v = ((const float4*)src)[i];
        v.x *= s; v.y *= s; v.z *= s; v.w *= s;
        ((float4*)dst)[i] = v;
    }
}

__global__ __launch_bounds__(256) void cheb_bias(float* __restrict__ out,
                                                 const float* __restrict__ bias, int n) {
    int i = blockIdx.x * 256 + threadIdx.x;
    if (i < n) out[i] = bias[i & (OC - 1)];
}

// ---------------- graph preprocessing ----------------

__global__ __launch_bounds__(256) void cheb_degree(const int* __restrict__ rows,
                                                   const int* __restrict__ cols,
                                                   float* __restrict__ deg, int E) {
    int e = blockIdx.x * 256 + threadIdx.x;
    if (e < E) {
        int r = rows[e];
        if (r != cols[e]) atomicAdd(&deg[r], 1.0f);
    }
}

__global__ __launch_bounds__(256) void cheb_rsqrt(float* __restrict__ deg, int n) {
    int i = blockIdx.x * 256 + threadIdx.x;
    if (i < n) {
        float d = deg[i];
        deg[i] = (d > 0.f) ? rsqrtf(d) : 0.f;
    }
}

__global__ __launch_bounds__(256) void cheb_lap(const int* __restrict__ rows,
                                                const int* __restrict__ cols,
                                                const float* __restrict__ ew,
                                                const float* __restrict__ dinv,
                                                float* __restrict__ lap, int E) {
    int e = blockIdx.x * 256 + threadIdx.x;
    if (e < E) {
        int r = rows[e], c = cols[e];
        lap[e] = (r != c) ? (-dinv[r] * ew[e] * dinv[c]) : 0.f;
    }
}

// ---------------- SpMM scatter: dst[row,:] += coef*lap[e]*src[col,:] ----------------
// 64 lanes per edge, 4 channels per lane (float4 gather, 4x global_atomic_add_f32).

__global__ __launch_bounds__(256) void cheb_spmm(float* __restrict__ dst,
                                                 const float* __restrict__ src,
                                                 const float* __restrict__ lap,
                                                 const int* __restrict__ rows,
                                                 const int* __restrict__ cols,
                                                 float coef, int E) {
    int t = blockIdx.x * 256 + threadIdx.x;
    int e = t >> 6;
    if (e >= E) return;
    float w = coef * lap[e];
    if (w == 0.f) return;
    int c4 = (t & 63) << 2;
    int r = rows[e], c = cols[e];
    const float4 v = *(const float4*)(src + (size_t)c * IC + c4);
    float* o = dst + (size_t)r * IC + c4;
    atomicAdd(o + 0, w * v.x);
    atomicAdd(o + 1, w * v.y);
    atomicAdd(o + 2, w * v.z);
    atomicAdd(o + 3, w * v.w);
}

// ---------------- dense GEMM: out += T @ W  (T: N x 256, W: 256 x 256) ----------------

#ifdef HAVE_WMMA_F32
// One wave per 16x16 output tile; 8 waves per block = 128 rows per block.
// V_WMMA_F32_16X16X4_F32 fragment layout (wave32):
//   A (16x4):  lane<16 -> M=lane, {K=kk, kk+1}; lane>=16 -> M=lane-16, {K=kk+2, kk+3}
//   B (4x16):  lane<16 -> N=lane, rows {kk, kk+1}; lane>=16 -> N=lane-16, rows {kk+2, kk+3}
//   C (16x16): vgpr r: lane<16 -> (M=r, N=lane); lane>=16 -> (M=r+8, N=lane-16)
__global__ __launch_bounds__(256) void cheb_gemm(const float* __restrict__ T,
                                                 const float* __restrict__ W,
                                                 float* __restrict__ out, int N) {
    const int lane = threadIdx.x & 31;
    const int wave = threadIdx.x >> 5;
    const int m0 = (blockIdx.x * 8 + wave) * 16;
    if (m0 >= N) return;                 // uniform per wave: EXEC stays all-ones for WMMA
    const int n0 = blockIdx.y * 16;
    const int half = lane >> 4;          // 0 or 1
    const int l = lane & 15;

    // load C accumulator tile from out
    v8f c;
    {
        const float* cp = out + (size_t)(m0 + half * 8) * OC + n0 + l;
#pragma unroll
        for (int r = 0; r < 8; ++r) c[r] = cp[(size_t)r * OC];
    }

    const float* arow = T + (size_t)(m0 + l) * IC + half * 2;   // 8B-aligned v2f loads
    const float* bp   = W + (size_t)(half * 2) * OC + n0 + l;

#pragma unroll 4
    for (int kk = 0; kk < IC; kk += 4) {
        v2f a = *(const v2f*)(arow + kk);
        v2f b;
        b.x = bp[(size_t)kk * OC];
        b.y = bp[(size_t)kk * OC + OC];
        c = __builtin_amdgcn_wmma_f32_16x16x4_f32(false, a, false, b,
                                                  (short)0, c, false, false);
    }

    float* cp = out + (size_t)(m0 + half * 8) * OC + n0 + l;
#pragma unroll
    for (int r = 0; r < 8; ++r) cp[(size_t)r * OC] = c[r];
}
#else
// Scalar fallback (only used if the f32 WMMA builtin is unavailable).
__global__ __launch_bounds__(256) void cheb_gemm(const float* __restrict__ T,
                                                 const float* __restrict__ W,
                                                 float* __restrict__ out, int N) {
    int i = blockIdx.x * 256 + threadIdx.x;   // launched with same grid: reinterpret
    int m = blockIdx.x * 2 + (threadIdx.x >> 7);       // 128 rows per block.x? keep simple:
    (void)i;
    m = blockIdx.x * 8 * 16 + (threadIdx.x >> 4);      // 16 threads share a row? -> guard
    if (m >= N) return;
    int n = blockIdx.y * 16 + (threadIdx.x & 15);
    float acc = out[(size_t)m * OC + n];
    const float* tr = T + (size_t)m * IC;
    for (int k = 0; k < IC; ++k) acc += tr[k] * W[(size_t)k * OC + n];
    out[(size_t)m * OC + n] = acc;
}
#endif

// ---------------- launcher ----------------

extern "C" void kernel_launch(void* const* d_in, const int* in_sizes, int n_in,
                              void* d_out, int out_size, void* d_ws, size_t ws_size,
                              hipStream_t stream) {
    const float* x    = (const float*)d_in[0];
    const int*   ei   = (const int*)d_in[1];
    const float* ew   = (const float*)d_in[2];
    const float* wgt  = (const float*)d_in[3];
    const float* bias = (const float*)d_in[4];
    float* out = (float*)d_out;

    const int N = in_sizes[0] / IC;
    const int E = in_sizes[1] / 2;
    const int K = in_sizes[3] / (IC * OC);

    const int* rows = ei;
    const int* cols = ei + E;

    // workspace layout
    float* deg  = (float*)d_ws;
    float* lap  = deg + ((N + 63) & ~63);
    float* bufA = lap + ((E + 63) & ~63);
    float* bufB = bufA + (size_t)N * IC;

    const int nfeat = N * IC;
    const int gN   = (N + 255) / 256;
    const int gE   = (E + 255) / 256;
    const int gF   = (nfeat + 255) / 256;
    const int gF4  = (nfeat / 4 + 255) / 256;
    const int gSp  = (E * 64 + 255) / 256;
    const int mtiles = (N + 15) / 16;
    dim3 ggrid((mtiles + 7) / 8, OC / 16);

    // 1) degree -> D^{-1/2} -> lap
    cheb_zero  <<<gN, 256, 0, stream>>>(deg, N);
    cheb_degree<<<gE, 256, 0, stream>>>(rows, cols, deg, E);
    cheb_rsqrt <<<gN, 256, 0, stream>>>(deg, N);
    cheb_lap   <<<gE, 256, 0, stream>>>(rows, cols, ew, deg, lap, E);

    // 2) out = bias; out += x @ W0
    cheb_bias<<<gF, 256, 0, stream>>>(out, bias, nfeat);
    cheb_gemm<<<ggrid, 256, 0, stream>>>(x, wgt, out, N);

    if (K > 1) {
        // T1 = L x
        cheb_zero<<<gF, 256, 0, stream>>>(bufA, nfeat);
        cheb_spmm<<<gSp, 256, 0, stream>>>(bufA, x, lap, rows, cols, 1.0f, E);
        cheb_gemm<<<ggrid, 256, 0, stream>>>(bufA, wgt + (size_t)IC * OC, out, N);

        const float* Tm1 = bufA;   // T_{k-1}
        const float* Tm2 = x;      // T_{k-2}
        for (int k = 2; k < K; ++k) {
            // dst := -T_{k-2}; then dst += 2 * L * T_{k-1}  ==>  dst = T_k
            float* dst = (k == 2) ? bufB : (float*)Tm2;   // Tm2 lives in ws for k>2
            cheb_scale_copy<<<gF4, 256, 0, stream>>>(dst, Tm2, -1.0f, nfeat / 4);
            cheb_spmm<<<gSp, 256, 0, stream>>>(dst, Tm1, lap, rows, cols, 2.0f, E);
            cheb_gemm<<<ggrid, 256, 0, stream>>>(dst, wgt + (size_t)k * IC * OC, out, N);
            Tm2 = Tm1;
            Tm1 = dst;
        }
    }
}